// AnchorRadar_9921374454083
// MI455X (gfx1250) — compile-verified
//
#include <hip/hip_runtime.h>
#include <hip/hip_bf16.h>

typedef __attribute__((ext_vector_type(2))) float v2f;
typedef __attribute__((ext_vector_type(8))) float v8f;

#define EMB 64

// ---------------------------------------------------------------------------
// acc = x * (1/3); cur = x      (x = concat(featA, featB) row-major [N,64])
// float4-vectorized; all element counts are multiples of 4.
// ---------------------------------------------------------------------------
__global__ void lg_init_kernel(const float4* __restrict__ fa, long long nA4,
                               const float4* __restrict__ fb, long long nTot4,
                               float4* __restrict__ cur, float4* __restrict__ acc) {
    long long i = (long long)blockIdx.x * blockDim.x + threadIdx.x;
    if (i >= nTot4) return;
    float4 v = (i < nA4) ? fa[i] : fb[i - nA4];
    cur[i] = v;
    float4 a;
    a.x = v.x * (1.0f / 3.0f); a.y = v.y * (1.0f / 3.0f);
    a.z = v.z * (1.0f / 3.0f); a.w = v.w * (1.0f / 3.0f);
    acc[i] = a;
}

__global__ void lg_zero_kernel(float4* __restrict__ p, long long n4) {
    long long i = (long long)blockIdx.x * blockDim.x + threadIdx.x;
    if (i < n4) p[i] = make_float4(0.f, 0.f, 0.f, 0.f);
}

// ---------------------------------------------------------------------------
// Edge-parallel SpMM scatter: one thread per (edge, dim).
// Gather cur[col*64 + d] is contiguous across d  -> fully coalesced 128B/wave.
// Accumulate with global_atomic_add_f32 (resolves in L2; nxt fits in 192MB L2).
// ---------------------------------------------------------------------------
__global__ void lg_scatter_kernel(const int* __restrict__ rows,
                                  const int* __restrict__ cols,
                                  const float* __restrict__ vals,
                                  const float* __restrict__ cur,
                                  float* __restrict__ nxt,
                                  long long nWork, long long nE) {
    long long idx = (long long)blockIdx.x * blockDim.x + threadIdx.x;
    if (idx >= nWork) return;
    long long e = idx >> 6;          // edge entry
    int       d = (int)(idx & 63);   // embedding dim
    int r = rows[e];
    int c = cols[e];
    float v = vals[e];
    // speculative prefetch of future index data (global_prefetch_b8)
    if (e + 4096 < nE) __builtin_prefetch(&cols[e + 4096], 0, 0);
    float g = cur[(size_t)c * EMB + d];
    atomicAdd(&nxt[(size_t)r * EMB + d], v * g);
}

// ---------------------------------------------------------------------------
// WMMA norm + accumulate: one wave32 per 16-row tile.
//   G = X * X^T via 16 chained V_WMMA_F32_16X16X4_F32 (K=64).
//   f32 A(16x4) and B(4x16) lane layouts are duals -> same VGPRs feed A and B:
//     lane L holds float2 of row (L&15) at col 4k + 2*(L>>4), k = 0..15.
//   diag(G)[m] = ||row m||^2 at: vgpr m&7, lane (m<8 ? m : m+16).
//   acc[row] += nxt[row] * (1/3) / max(||row||, 1e-12)
// N is divisible by 16 for all three graphs, so EXEC is all-ones at each WMMA.
// ---------------------------------------------------------------------------
__global__ void lg_norm_acc_kernel(const float* __restrict__ x,
                                   float* __restrict__ acc,
                                   int nTiles) {
    __shared__ float diag[8][16];
    const int lane = threadIdx.x & 31;
    const int wave = threadIdx.x >> 5;
    const int tile = blockIdx.x * (blockDim.x >> 5) + wave;
    if (tile >= nTiles) return;          // wave-uniform; EXEC stays full

    const int r = lane & 15;             // row within tile
    const int h = lane >> 4;             // half-wave -> K sub-pair
    const float* rowp = x + ((size_t)tile * 16 + r) * EMB + 2 * h;

    v2f a[16];
#pragma unroll
    for (int k = 0; k < 16; ++k)
        a[k] = *(const v2f*)(rowp + 4 * k);

    v8f c = {};
#pragma unroll
    for (int k = 0; k < 16; ++k)
        c = __builtin_amdgcn_wmma_f32_16x16x4_f32(
                /*neg_a=*/false, a[k], /*neg_b=*/false, a[k],
                /*c_mod=*/(short)0, c, /*reuse_a=*/false, /*reuse_b=*/false);

    // scatter diagonal of G into per-wave LDS
#pragma unroll
    for (int j = 0; j < 8; ++j) {
        if (lane == j)      diag[wave][j]     = c[j];
        if (lane == j + 24) diag[wave][j + 8] = c[j];
    }
    __builtin_amdgcn_wave_barrier();     // DS ops are in-order within a wave
    float s   = diag[wave][r];
    float nrm = sqrtf(s);
    float inv = (1.0f / 3.0f) / fmaxf(nrm, 1e-12f);

    float* accp = acc + ((size_t)tile * 16 + r) * EMB + 2 * h;
#pragma unroll
    for (int k = 0; k < 16; ++k) {
        v2f o = *(v2f*)(accp + 4 * k);
        o.x += a[k].x * inv;
        o.y += a[k].y * inv;
        *(v2f*)(accp + 4 * k) = o;
    }
}

// ---------------------------------------------------------------------------
// Host-side orchestration
// ---------------------------------------------------------------------------
static void run_graph(const float* fa, int nA, const float* fb, int nB,
                      const int* er, const int* ec, const float* ev, long long nE,
                      float* acc, float* bufA, float* bufB, hipStream_t stream) {
    const int N = nA + nB;
    const long long tot  = (long long)N * EMB;
    const long long tot4 = tot / 4;
    const long long nA4  = (long long)nA * EMB / 4;

    lg_init_kernel<<<(unsigned)((tot4 + 255) / 256), 256, 0, stream>>>(
        (const float4*)fa, nA4, (const float4*)fb, tot4,
        (float4*)bufA, (float4*)acc);

    float* cur = bufA;
    float* nxt = bufB;
    for (int layer = 0; layer < 2; ++layer) {
        lg_zero_kernel<<<(unsigned)((tot4 + 255) / 256), 256, 0, stream>>>(
            (float4*)nxt, tot4);

        long long nWork = nE * EMB;
        lg_scatter_kernel<<<(unsigned)((nWork + 255) / 256), 256, 0, stream>>>(
            er, ec, ev, cur, nxt, nWork, nE);

        int nTiles = N / 16;  // all graph sizes are multiples of 16
        lg_norm_acc_kernel<<<(unsigned)((nTiles + 7) / 8), 256, 0, stream>>>(
            nxt, acc, nTiles);

        float* t = cur; cur = nxt; nxt = t;
    }
}

extern "C" void kernel_launch(void* const* d_in, const int* in_sizes, int n_in,
                              void* d_out, int out_size, void* d_ws, size_t ws_size,
                              hipStream_t stream) {
    const float* fu = (const float*)d_in[0];
    const float* fb = (const float*)d_in[1];
    const float* fi = (const float*)d_in[2];

    const int* ub_r = (const int*)d_in[3];
    const int* ub_c = (const int*)d_in[4];
    const float* ub_v = (const float*)d_in[5];
    const long long nE_ub = in_sizes[3];

    const int* ui_r = (const int*)d_in[6];
    const int* ui_c = (const int*)d_in[7];
    const float* ui_v = (const float*)d_in[8];
    const long long nE_ui = in_sizes[6];

    const int* bi_r = (const int*)d_in[9];
    const int* bi_c = (const int*)d_in[10];
    const float* bi_v = (const float*)d_in[11];
    const long long nE_bi = in_sizes[9];

    const int NU = in_sizes[0] / EMB;   // 50000
    const int NB = in_sizes[1] / EMB;   // 20000
    const int NI = in_sizes[2] / EMB;   // 100000

    float* out = (float*)d_out;
    float* acc_ub = out;                                    // [NU+NB, 64]
    float* acc_ui = acc_ub + (size_t)(NU + NB) * EMB;       // [NU+NI, 64]
    float* acc_bi = acc_ui + (size_t)(NU + NI) * EMB;       // [NB+NI, 64]

    // ping-pong feature buffers sized for the largest graph (NU+NI)
    size_t maxN = (size_t)(NU + NI);
    if ((size_t)(NU + NB) > maxN) maxN = NU + NB;
    if ((size_t)(NB + NI) > maxN) maxN = NB + NI;
    float* bufA = (float*)d_ws;
    float* bufB = bufA + maxN * EMB;

    run_graph(fu, NU, fb, NB, ub_r, ub_c, ub_v, nE_ub, acc_ub, bufA, bufB, stream);
    run_graph(fu, NU, fi, NI, ui_r, ui_c, ui_v, nE_ui, acc_ui, bufA, bufB, stream);
    run_graph(fb, NB, fi, NI, bi_r, bi_c, bi_v, nE_bi, acc_bi, bufA, bufB, stream);
}